// MaxPooling_49022756717276
// MI455X (gfx1250) — compile-verified
//
#include <hip/hip_runtime.h>

// Rule-book max pooling:  out[o, :] = max_k feats[rules[o, k], :]
//   feats : [N_IN  = 1e6,  C = 64]  fp32  (256 B per row, 256-B aligned)
//   rules : [N_OUT = 250e3, K = 8]  int32
//   out   : [N_OUT, C]              fp32
//
// Pure bandwidth kernel (~584 MB traffic -> ~25 us floor at 23.3 TB/s).
// - 16 lanes x float4 cover one 256-B feature row; one wave32 handles two
//   output rows per gather instruction (each global_load_b128 moves 512 B).
// - All rule rows for a wave's 4 iterations are staged to LDS up-front with
//   one GLOBAL_LOAD_ASYNC_TO_LDS_B128 per lane (ASYNCcnt-tracked), fully
//   decoupling index fetch from the dependent gather chain.
// - Feature gathers use default RT caching (2x reuse lives in the 192 MB L2);
//   output uses non-temporal b128 stores (written once, never re-read).

typedef float v4f __attribute__((ext_vector_type(4)));
typedef int   v4i __attribute__((ext_vector_type(4)));

#define KK 8
#define CC 64
#define WAVES_PER_BLOCK 8            // 256 threads = 8 wave32 on CDNA5
#define PAIR_ITERS 4                 // row-pairs per wave
#define THREADS (WAVES_PER_BLOCK * 32)
// rules staged per wave: PAIR_ITERS * 2 rows * 32 B = 256 B
#define RULE_INTS_PER_WAVE (PAIR_ITERS * 2 * KK)

static __device__ __forceinline__ v4f vmax4(v4f a, v4f b) {
  v4f r;
  r.x = fmaxf(a.x, b.x);
  r.y = fmaxf(a.y, b.y);
  r.z = fmaxf(a.z, b.z);
  r.w = fmaxf(a.w, b.w);
  return r;
}

__global__ __launch_bounds__(THREADS)
void maxpool_rule_gather(const float* __restrict__ feats,
                         const int*   __restrict__ rules,
                         float*       __restrict__ out,
                         int n_out, int n_pairs, int stride_pairs)
{
  __shared__ int lds_rules[WAVES_PER_BLOCK * RULE_INTS_PER_WAVE];  // 2 KB

  const int lane  = threadIdx.x & 31;
  const int wave  = threadIdx.x >> 5;
  const int sel   = lane >> 4;     // which row of the pair this half-wave owns
  const int half  = lane & 15;     // 16-lane channel group: 4 floats per lane
  const int pair0 = blockIdx.x * WAVES_PER_BLOCK + wave;

  int* wave_rules = lds_rules + wave * RULE_INTS_PER_WAVE;

  // ---- Stage this wave's 4 rule-row-pairs (256 B) into LDS asynchronously.
  // Lane l (0..15) covers iteration it = l>>2, 16-B quad q = l&3, landing at
  // LDS offset l*16 == it*64 + q*16, so iteration `it`, row `sel` later reads
  // ints [it*16 + sel*8, +8).
  if (lane < 16) {
    const int it   = lane >> 2;
    const int quad = lane & 3;
    int pr = pair0 + it * stride_pairs;
    if (pr >= n_pairs) pr = 0;     // clamp: async loads are not speculative
    const int* src = rules + (size_t)pr * (2 * KK) + quad * 4;
    // Generic->LDS address: hardware uses addr[31:0] (ISA 10.2), which equals
    // the ds-space offset the compiler uses for the same __shared__ object.
    unsigned lds_off = (unsigned)(uintptr_t)wave_rules + (unsigned)lane * 16u;
    asm volatile("global_load_async_to_lds_b128 %0, %1, off"
                 :: "v"(lds_off), "v"(src)
                 : "memory");
  }
  asm volatile("s_wait_asynccnt 0x0" ::: "memory");

  const v4f* __restrict__ f4 = (const v4f*)feats;

  int pair = pair0;
  for (int it = 0; it < PAIR_ITERS; ++it) {
    const int row = pair * 2 + sel;
    if (row < n_out) {
      // Indices from LDS (ds_load_b128 x2) -- no global round-trip here.
      const v4i* rr = (const v4i*)(wave_rules + it * 16 + sel * 8);
      v4i i0 = rr[0];
      v4i i1 = rr[1];

      // 8 independent b128 gathers in flight before the reduce.
      // Byte offsets max out at 256 MB, safely inside 32-bit voffset range.
      v4f g0 = f4[(unsigned)i0.x * 16u + half];
      v4f g1 = f4[(unsigned)i0.y * 16u + half];
      v4f g2 = f4[(unsigned)i0.z * 16u + half];
      v4f g3 = f4[(unsigned)i0.w * 16u + half];
      v4f g4 = f4[(unsigned)i1.x * 16u + half];
      v4f g5 = f4[(unsigned)i1.y * 16u + half];
      v4f g6 = f4[(unsigned)i1.z * 16u + half];
      v4f g7 = f4[(unsigned)i1.w * 16u + half];

      v4f m = vmax4(vmax4(vmax4(g0, g1), vmax4(g2, g3)),
                    vmax4(vmax4(g4, g5), vmax4(g6, g7)));

      // Output written once, never re-read -> non-temporal b128 store.
      __builtin_nontemporal_store(m, (v4f*)out + (size_t)row * (CC / 4) + half);
    }
    pair += stride_pairs;
  }
}

extern "C" void kernel_launch(void* const* d_in, const int* in_sizes, int n_in,
                              void* d_out, int out_size, void* d_ws, size_t ws_size,
                              hipStream_t stream) {
  const float* feats = (const float*)d_in[0];
  const int*   rules = (const int*)d_in[1];
  float*       out   = (float*)d_out;

  const int n_out   = out_size / CC;                          // 250,000
  const int n_pairs = (n_out + 1) / 2;                        // 2 rows per wave
  const int stride  = (n_pairs + PAIR_ITERS - 1) / PAIR_ITERS; // waves needed
  const int blocks  = (stride + WAVES_PER_BLOCK - 1) / WAVES_PER_BLOCK;

  maxpool_rule_gather<<<blocks, THREADS, 0, stream>>>(
      feats, rules, out, n_out, n_pairs, stride);
}